// SeriesDecomp_5437428597105
// MI455X (gfx1250) — compile-verified
//
#include <hip/hip_runtime.h>
#include <math.h>

// ---------------------------------------------------------------------------
// SeriesDecomp on MI455X (gfx1250): exact fp32 spectral projection.
//   seasonal = (1/N) Sum_k w_k m_k (C_k cos(2pi k t/N) + S_k sin(2pi k t/N))
//   trend    = x - seasonal_raw ;  blend on seasonal head (409 rows)
// Two fp32 WMMA GEMM stages over K=1024 bins (mask < 2e-8 beyond).
// Basis from an 8192-entry LDS cos table (two periods, 32KB):
//   - loop-carried BYTE offsets masked to 0x3FFC (no lshl per use)
//   - sin(th)=cos(th-pi/2) is a constant +12288B ds offset (no wrap math)
// ---------------------------------------------------------------------------

typedef float v2f __attribute__((ext_vector_type(2)));
typedef float v8f __attribute__((ext_vector_type(8)));

#define N_T 4096
#define N_D 512
#define N_B 8
#define K_BINS 1024
#define BLEND_LEN 409
#define TWO_PI_OVER_N (6.28318530717958647692f / 4096.0f)
#define MSK 16380        // (4096-1)*4 byte mask, keeps offsets DWORD-aligned
#define SINOFF 12288     // 3072 elements * 4 bytes

#define WMMA_F32(A, B, C) \
  __builtin_amdgcn_wmma_f32_16x16x4_f32(false, (A), false, (B), (short)0, (C), false, false)

// ---- setup: per-bin scale = min(Gaussians,1) * w_k / N --------------------
__global__ void SD_scale_kernel(float* __restrict__ scaletab) {
  int k = blockIdx.x * blockDim.x + threadIdx.x;
  if (k >= K_BINS) return;
  float f = (float)k / (float)N_T;
  const float pf = 1.0f / 24.0f;
  const float inv_sigma = 48.0f;             // sigma = 1/(period*2) = 1/48
  float m = 0.0f;
  {
    float u = (f - pf) * inv_sigma;          // seasonal fundamental
    m += expf(-0.5f * u * u);
  }
  for (int h = 1; h <= 3; ++h) {             // harmonics (h=1 duplicates)
    float u = (f - pf * (float)h) * inv_sigma;
    m += expf(-0.5f * u * u);
  }
  m = fminf(m, 1.0f);
  float w = (k == 0) ? 1.0f : 2.0f;
  scaletab[k] = m * w / (float)N_T;
}

__device__ __forceinline__ float lds_at(const float* tab, int byte_off) {
  return *(const float*)((const char*)tab + byte_off);
}

// ---- stage 1: Y_{c,s}[b,k,d] = Sum_t x * {cos,sin}, scaled ----------------
// wave tile: 16 kbins x 32 d (two 16x16 WMMA tiles sharing the A basis).
__global__ __launch_bounds__(256) void SD_fwd_kernel(
    const float* __restrict__ x, const float* __restrict__ scaletab,
    float* __restrict__ yc, float* __restrict__ ys) {
  __shared__ float costab[8192];             // two periods, 32KB
  for (int i = threadIdx.x; i < 8192; i += 256)
    costab[i] = cosf(TWO_PI_OVER_N * (float)i);
  __syncthreads();

  const int wave = (int)((blockIdx.x * blockDim.x + threadIdx.x) >> 5);
  const int lane = (int)(threadIdx.x & 31);
  const int dp = wave & 15;                  // d pair   (16)
  const int kt = (wave >> 4) & 63;           // k tile   (64)
  const int b  = wave >> 10;                 // batch    (8)
  const int li = lane & 15;
  const int lh = lane >> 4;
  const int koff = lh * 2;                   // K-pair owned by this lane half

  const int krow = kt * 16 + li;             // A row (M = frequency bin)
  const int d0 = dp * 32 + li;               // B col, tile 0
  const int d1 = d0 + 16;                    // B col, tile 1
  const float* xb = x + (size_t)b * N_T * N_D;

  v8f cc0 = {}, cs0 = {}, cc1 = {}, cs1 = {};
  int m0b = ((krow * koff) & (N_T - 1)) * 4;       // (k*t mod N)*4, t=koff
  int m1b = ((krow * (koff + 1)) & (N_T - 1)) * 4; // t=koff+1
  const int stb = ((krow * 4) & (N_T - 1)) * 4;    // per-step advance

  for (int s = 0; s < N_T; s += 4) {
    v2f a, sa, b0, b1;
    a.x  = lds_at(costab, m0b);
    a.y  = lds_at(costab, m1b);
    sa.x = lds_at(costab, m0b + SINOFF);     // folds into ds offset
    sa.y = lds_at(costab, m1b + SINOFF);
    const float* xr = xb + (size_t)(s + koff) * N_D;
    b0.x = xr[d0];  b0.y = xr[N_D + d0];
    b1.x = xr[d1];  b1.y = xr[N_D + d1];
    cc0 = WMMA_F32(a,  b0, cc0);
    cc1 = WMMA_F32(a,  b1, cc1);
    cs0 = WMMA_F32(sa, b0, cs0);
    cs1 = WMMA_F32(sa, b1, cs1);
    m0b = (m0b + stb) & MSK;
    m1b = (m1b + stb) & MSK;
  }
#pragma unroll
  for (int v = 0; v < 8; ++v) {
    const int k = kt * 16 + v + lh * 8;      // C/D layout: M = v + 8*lanehalf
    const float sc = scaletab[k];
    const size_t row = ((size_t)b * K_BINS + k) * N_D;
    yc[row + d0] = cc0[v] * sc;
    ys[row + d0] = cs0[v] * sc;
    yc[row + d1] = cc1[v] * sc;
    ys[row + d1] = cs1[v] * sc;
  }
}

// ---- stage 2: seasonal = Sum_k Yc cos + Ys sin ; trend = x - seasonal -----
// wave tile: 32 t x 32 d (2x2 WMMA tiles; A shared over d, B shared over t).
__global__ __launch_bounds__(256) void SD_inv_kernel(
    const float* __restrict__ x, const float* __restrict__ yc,
    const float* __restrict__ ys, float* __restrict__ trend,
    float* __restrict__ seasonal) {
  __shared__ float costab[8192];
  for (int i = threadIdx.x; i < 8192; i += 256)
    costab[i] = cosf(TWO_PI_OVER_N * (float)i);
  __syncthreads();

  const int wave = (int)((blockIdx.x * blockDim.x + threadIdx.x) >> 5);
  const int lane = (int)(threadIdx.x & 31);
  const int dp = wave & 15;                  // d pair  (16)
  const int tp = (wave >> 4) & 127;          // t pair  (128)
  const int b  = wave >> 11;                 // batch   (8)
  const int li = lane & 15;
  const int lh = lane >> 4;
  const int koff = lh * 2;

  const int t0 = tp * 32 + li;               // A row, tile 0 (M = time)
  const int t1 = t0 + 16;                    // A row, tile 1
  const int d0 = dp * 32 + li;
  const int d1 = d0 + 16;
  const float* ycb = yc + (size_t)b * K_BINS * N_D;
  const float* ysb = ys + (size_t)b * K_BINS * N_D;

  v8f c00 = {}, c01 = {}, c10 = {}, c11 = {};
  int a0b = ((t0 * koff) & (N_T - 1)) * 4;         // (t*k mod N)*4, k=koff
  int a1b = ((t0 * (koff + 1)) & (N_T - 1)) * 4;
  int b0b = ((t1 * koff) & (N_T - 1)) * 4;
  int b1b = ((t1 * (koff + 1)) & (N_T - 1)) * 4;
  const int st0 = ((t0 * 4) & (N_T - 1)) * 4;
  const int st1 = ((t1 * 4) & (N_T - 1)) * 4;

  for (int s = 0; s < K_BINS; s += 4) {
    v2f a0, sa0, a1, sa1, bc0, bc1, bs0, bs1;
    a0.x  = lds_at(costab, a0b);
    a0.y  = lds_at(costab, a1b);
    sa0.x = lds_at(costab, a0b + SINOFF);
    sa0.y = lds_at(costab, a1b + SINOFF);
    a1.x  = lds_at(costab, b0b);
    a1.y  = lds_at(costab, b1b);
    sa1.x = lds_at(costab, b0b + SINOFF);
    sa1.y = lds_at(costab, b1b + SINOFF);
    const float* pc = ycb + (size_t)(s + koff) * N_D;
    const float* ps = ysb + (size_t)(s + koff) * N_D;
    bc0.x = pc[d0];  bc0.y = pc[N_D + d0];
    bc1.x = pc[d1];  bc1.y = pc[N_D + d1];
    bs0.x = ps[d0];  bs0.y = ps[N_D + d0];
    bs1.x = ps[d1];  bs1.y = ps[N_D + d1];
    c00 = WMMA_F32(a0,  bc0, c00);
    c01 = WMMA_F32(a0,  bc1, c01);
    c10 = WMMA_F32(a1,  bc0, c10);
    c11 = WMMA_F32(a1,  bc1, c11);
    c00 = WMMA_F32(sa0, bs0, c00);
    c01 = WMMA_F32(sa0, bs1, c01);
    c10 = WMMA_F32(sa1, bs0, c10);
    c11 = WMMA_F32(sa1, bs1, c11);
    a0b = (a0b + st0) & MSK;
    a1b = (a1b + st0) & MSK;
    b0b = (b0b + st1) & MSK;
    b1b = (b1b + st1) & MSK;
  }
  const float* xb = x + (size_t)b * N_T * N_D;
#pragma unroll
  for (int v = 0; v < 8; ++v) {
    const int ta = tp * 32 + v + lh * 8;     // tile-0 row
    const int tb = ta + 16;                  // tile-1 row
    const size_t ra = ((size_t)b * N_T + ta) * N_D;
    const size_t rb = ((size_t)b * N_T + tb) * N_D;
    float s00 = c00[v], s01 = c01[v], s10 = c10[v], s11 = c11[v];
    trend[ra + d0] = xb[(size_t)ta * N_D + d0] - s00;
    trend[ra + d1] = xb[(size_t)ta * N_D + d1] - s01;
    trend[rb + d0] = xb[(size_t)tb * N_D + d0] - s10;
    trend[rb + d1] = xb[(size_t)tb * N_D + d1] - s11;
    seasonal[ra + d0] = s00;
    seasonal[ra + d1] = s01;
    seasonal[rb + d0] = s10;
    seasonal[rb + d1] = s11;
  }
}

// ---- stage 3: in-place circular blend on seasonal head --------------------
// writes only t < 409; reads t and N-1-t (>= 3687): disjoint, deterministic.
__global__ void SD_blend_kernel(float* __restrict__ seasonal) {
  const int total = N_B * BLEND_LEN * N_D;
  int i = blockIdx.x * blockDim.x + threadIdx.x;
  if (i >= total) return;
  const int d = i % N_D;
  const int t = (i / N_D) % BLEND_LEN;
  const int b = i / (N_D * BLEND_LEN);
  const float r = (float)t / (float)(BLEND_LEN - 1);
  const size_t ih = ((size_t)b * N_T + t) * N_D + d;
  const size_t im = ((size_t)b * N_T + (N_T - 1 - t)) * N_D + d;
  const float head = seasonal[ih];
  const float tail = seasonal[im];
  seasonal[ih] = head * r + tail * (1.0f - r);
}

extern "C" void kernel_launch(void* const* d_in, const int* in_sizes, int n_in,
                              void* d_out, int out_size, void* d_ws, size_t ws_size,
                              hipStream_t stream) {
  (void)in_sizes; (void)n_in; (void)out_size; (void)ws_size;
  const float* x = (const float*)d_in[0];
  float* trend    = (float*)d_out;                            // [8,4096,512]
  float* seasonal = (float*)d_out + (size_t)N_B * N_T * N_D;  // [8,4096,512]

  // ws: scaletab[1024] | Yc[8,1024,512] | Ys[8,1024,512]  (~32MB)
  float* scaletab = (float*)d_ws;
  float* yc = (float*)((char*)d_ws + 4096);
  float* ys = yc + (size_t)N_B * K_BINS * N_D;

  SD_scale_kernel<<<(K_BINS + 255) / 256, 256, 0, stream>>>(scaletab);

  // stage 1: 8 b * 64 ktiles * 16 dpairs = 8192 waves, 8 waves/block
  SD_fwd_kernel<<<1024, 256, 0, stream>>>(x, scaletab, yc, ys);

  // stage 2: 8 b * 128 tpairs * 16 dpairs = 16384 waves, 8 waves/block
  SD_inv_kernel<<<2048, 256, 0, stream>>>(x, yc, ys, trend, seasonal);

  const int btotal = N_B * BLEND_LEN * N_D;
  SD_blend_kernel<<<(btotal + 255) / 256, 256, 0, stream>>>(seasonal);
}